// OptFP_Embedding_73426760892790
// MI455X (gfx1250) — compile-verified
//
#include <hip/hip_runtime.h>
#include <cstdint>
#include <cstddef>

// ---------------------------------------------------------------------------
// OptFP embedding fake-quant gather kernel for MI455X (gfx1250).
// Memory-bound: ~83 MB moved, ~3.6 us roofline @ 23.3 TB/s. No matmul exists
// in this op, so the CDNA5 path exercised here is the async-to-LDS gather
// pipeline (ASYNCcnt) + b128 vector memory + NT streamed stores, not WMMA.
// ---------------------------------------------------------------------------

typedef float v4f __attribute__((ext_vector_type(4)));
typedef int   v4i __attribute__((ext_vector_type(4)));

#ifndef __has_builtin
#define __has_builtin(x) 0
#endif

#if __has_builtin(__builtin_amdgcn_global_load_async_to_lds_b128)
#define OPTFP_HAVE_ASYNC 1
#else
#define OPTFP_HAVE_ASYNC 0
#endif

#if OPTFP_HAVE_ASYNC
// Builtin param types (from clang diagnostic): (int4 AS1*, int4 AS3*, imm, imm).
// Generic pointers are laundered through integers to reach the target address
// spaces (generic->AS1/AS3 truncation matches the hardware addrspacecast).
__device__ __forceinline__ void optfp_async_gather_b128(const void* gsrc, void* ldst) {
  __builtin_amdgcn_global_load_async_to_lds_b128(
      (__attribute__((address_space(1))) v4i*)(uint64_t)(uintptr_t)gsrc,
      (__attribute__((address_space(3))) v4i*)(uint32_t)(uintptr_t)ldst,
      0, 0);
}
#endif

__device__ __forceinline__ void optfp_wait_async(int allow_one_outstanding) {
#if OPTFP_HAVE_ASYNC
#if __has_builtin(__builtin_amdgcn_s_wait_asynccnt)
  if (allow_one_outstanding) __builtin_amdgcn_s_wait_asynccnt(1);
  else                       __builtin_amdgcn_s_wait_asynccnt(0);
#else
  if (allow_one_outstanding) asm volatile("s_wait_asynccnt 1" ::: "memory");
  else                       asm volatile("s_wait_asynccnt 0" ::: "memory");
#endif
  asm volatile("" ::: "memory");  // compiler barrier: don't hoist LDS reads above the wait
#else
  (void)allow_one_outstanding;
#endif
}

constexpr int kThreads       = 256;  // 8 waves (wave32) per block
constexpr int kLanesPerTok   = 16;   // 16 lanes x float4 = 64-float row
constexpr int kToksPerTile   = kThreads / kLanesPerTok;  // 16 tokens per tile
constexpr int kTilesPerBlock = 4;    // software-pipelined tiles per block
constexpr int kEmb           = 64;

// out = sum_i gm_i * ( clip(rint((w-beta)*inv_a_i), lo_i, hi_i) * a_i + beta )
// BITSETS = (2,4,8) -> clip ranges [-2,1], [-8,7], [-128,127]
__device__ __forceinline__ v4f optfp_quant_mix(v4f w, v4f b4,
                                               float a0, float a1, float a2,
                                               float ia0, float ia1, float ia2,
                                               float gm0, float gm1, float gm2) {
  v4f o;
#pragma unroll
  for (int c = 0; c < 4; ++c) {
    float wv = w[c];
    float bv = b4[c];
    float d  = wv - bv;
    float r0 = fminf(fmaxf(rintf(d * ia0),   -2.0f),   1.0f);
    float r1 = fminf(fmaxf(rintf(d * ia1),   -8.0f),   7.0f);
    float r2 = fminf(fmaxf(rintf(d * ia2), -128.0f), 127.0f);
    float acc = fmaf(r0, a0, bv) * gm0;
    acc = fmaf(fmaf(r1, a1, bv), gm1, acc);
    acc = fmaf(fmaf(r2, a2, bv), gm2, acc);
    o[c] = acc;
  }
  return o;
}

__global__ __launch_bounds__(kThreads)
void optfp_embed_quant_kernel(const int*   __restrict__ x,       // [n_tok]
                              const float* __restrict__ weight,  // [1e6, 64]
                              const int*   __restrict__ gidx,    // [1e6]
                              const float* __restrict__ gamma,   // [8,1,3]
                              const float* __restrict__ alpha,   // [3]
                              const float* __restrict__ beta,    // [64]
                              float*       __restrict__ out,     // [n_tok, 64]
                              int n_tok, int n_tiles) {
  __shared__ float s_gm[8][4];   // softmax gates per group (padded row)
  __shared__ float s_a[4];
  __shared__ float s_ia[4];
#if OPTFP_HAVE_ASYNC
  __shared__ float s_w[2][kToksPerTile][kEmb];  // double-buffered gather staging (8 KB)
#endif

  const int tid = threadIdx.x;

  // Per-block scalar setup: 8-group x 3-bitwidth softmax table, |alpha|+eps, 1/a.
  if (tid < 8) {
    float g0 = gamma[tid * 3 + 0];
    float g1 = gamma[tid * 3 + 1];
    float g2 = gamma[tid * 3 + 2];   // TAU == 1.0
    float m  = fmaxf(g0, fmaxf(g1, g2));
    float e0 = __expf(g0 - m), e1 = __expf(g1 - m), e2 = __expf(g2 - m);
    float inv = 1.0f / (e0 + e1 + e2);
    s_gm[tid][0] = e0 * inv;
    s_gm[tid][1] = e1 * inv;
    s_gm[tid][2] = e2 * inv;
  }
  if (tid < 3) {
    float a = fabsf(alpha[tid]) + 1e-10f;
    s_a[tid]  = a;
    s_ia[tid] = 1.0f / a;
  }
  __syncthreads();

  const int sub  = tid >> 4;   // token-within-tile 0..15 (wave owns subs {2w,2w+1})
  const int lane = tid & 15;   // float4 slot within the 64-float row

  const v4f   b4  = ((const v4f*)beta)[lane];
  const float a0  = s_a[0],  a1  = s_a[1],  a2  = s_a[2];
  const float ia0 = s_ia[0], ia1 = s_ia[1], ia2 = s_ia[2];

  const int tile0 = blockIdx.x * kTilesPerBlock;

#if OPTFP_HAVE_ASYNC
  // -------- CDNA5 path: ASYNCcnt-pipelined gather via global_load_async_to_lds_b128
  int idx_cur = 0, g_cur = 0;
  if (tile0 < n_tiles) {
    int tok = tile0 * kToksPerTile + sub;
    int tc  = tok < n_tok ? tok : n_tok - 1;   // clamp: keep EXEC full, counters uniform
    idx_cur = x[tc];
    g_cur   = gidx[idx_cur];
    optfp_async_gather_b128(weight + (size_t)idx_cur * kEmb + lane * 4,
                            &s_w[0][sub][lane * 4]);
  }

  int buf = 0;
  for (int k = 0; k < kTilesPerBlock; ++k) {
    int t = tile0 + k;
    if (t >= n_tiles) break;
    const bool have_next = (k + 1 < kTilesPerBlock) && (t + 1 < n_tiles);

    int idx_nxt = 0, g_nxt = 0;
    if (have_next) {
      int tok = (t + 1) * kToksPerTile + sub;
      int tc  = tok < n_tok ? tok : n_tok - 1;
      idx_nxt = x[tc];
      g_nxt   = gidx[idx_nxt];
      optfp_async_gather_b128(weight + (size_t)idx_nxt * kEmb + lane * 4,
                              &s_w[buf ^ 1][sub][lane * 4]);
    }

    // Async loads complete in order: allowing 1 outstanding waits for tile t only.
    optfp_wait_async(have_next ? 1 : 0);

    v4f w = *(const v4f*)&s_w[buf][sub][lane * 4];
    float gm0 = s_gm[g_cur][0], gm1 = s_gm[g_cur][1], gm2 = s_gm[g_cur][2];
    v4f o = optfp_quant_mix(w, b4, a0, a1, a2, ia0, ia1, ia2, gm0, gm1, gm2);

    int tok = t * kToksPerTile + sub;
    if (tok < n_tok) {
      __builtin_nontemporal_store(o, (v4f*)(out + (size_t)tok * kEmb + lane * 4));
    }
    idx_cur = idx_nxt;
    g_cur   = g_nxt;
    buf ^= 1;
  }
#else
  // -------- Fallback: direct b128 gather loads (latency hidden by occupancy)
  for (int k = 0; k < kTilesPerBlock; ++k) {
    int t = tile0 + k;
    if (t >= n_tiles) break;
    int tok = t * kToksPerTile + sub;
    if (tok >= n_tok) continue;
    int idx = x[tok];
    int g   = gidx[idx];
    v4f w   = ((const v4f*)(weight + (size_t)idx * kEmb))[lane];
    float gm0 = s_gm[g][0], gm1 = s_gm[g][1], gm2 = s_gm[g][2];
    v4f o = optfp_quant_mix(w, b4, a0, a1, a2, ia0, ia1, ia2, gm0, gm1, gm2);
    __builtin_nontemporal_store(o, (v4f*)(out + (size_t)tok * kEmb + lane * 4));
  }
#endif
}

extern "C" void kernel_launch(void* const* d_in, const int* in_sizes, int n_in,
                              void* d_out, int out_size, void* d_ws, size_t ws_size,
                              hipStream_t stream) {
  (void)n_in; (void)out_size; (void)d_ws; (void)ws_size;
  const int*   x      = (const int*)d_in[0];     // [B*F] int32
  const float* weight = (const float*)d_in[1];   // [1e6,64] f32
  const int*   gidx   = (const int*)d_in[2];     // [1e6] int32
  const float* gamma  = (const float*)d_in[3];   // [8,1,3] f32
  const float* alpha  = (const float*)d_in[4];   // [3] f32
  const float* beta   = (const float*)d_in[5];   // [64] f32
  float*       out    = (float*)d_out;           // [B*F,64] f32

  const int n_tok   = in_sizes[0];
  const int n_tiles = (n_tok + kToksPerTile - 1) / kToksPerTile;
  const int blocks  = (n_tiles + kTilesPerBlock - 1) / kTilesPerBlock;

  optfp_embed_quant_kernel<<<blocks, kThreads, 0, stream>>>(
      x, weight, gidx, gamma, alpha, beta, out, n_tok, n_tiles);
}